// QuantLinearSim_13537736917852
// MI455X (gfx1250) — compile-verified
//
#include <hip/hip_runtime.h>
#include <hip/hip_bf16.h>
#include <math.h>

typedef __attribute__((ext_vector_type(2))) float v2f;
typedef __attribute__((ext_vector_type(4))) float v4f;
typedef __attribute__((ext_vector_type(8))) float v8f;
typedef __attribute__((ext_vector_type(4))) unsigned int v4u;
typedef __attribute__((ext_vector_type(4))) int v4i;
typedef __attribute__((ext_vector_type(8))) int v8i;

#define D_MODEL 4096
#define NOUT    4096
#define NTOK    2048   // B*S = 2*1024
#define MT      64     // macro-tile rows   (4 wave-rows x 16)
#define NT      128    // macro-tile cols   (2 wave-cols x 64)
#define KC      32     // K chunk staged in LDS
#define NPOLES  16
#define EPSV    1e-8f

#if __has_builtin(__builtin_amdgcn_tensor_load_to_lds)
#define USE_TDM 1
#endif

#ifdef USE_TDM
// Issue a 2-D TDM tile load: tile (tile1 rows x tile0 elems) of f32, row
// stride `stride0` elements, from global address `ga` into LDS offset
// `lds_off`. Descriptor per CDNA5 ISA ch.8 (D# groups 0/1; groups 2/3 zero).
// This toolchain exposes the 6-arg builtin: (v4u, v8i, v4i, v4i, v8i, cpol).
__device__ __forceinline__ void tdm_load_tile_f32(
    unsigned lds_off, unsigned long long ga,
    unsigned tdim0, unsigned tdim1, unsigned tile0, unsigned tile1,
    unsigned stride0) {
  v4u g0;
  g0.x = 1u;                                   // count=1, is_restore=0
  g0.y = lds_off;                              // lds_addr (bytes)
  g0.z = (unsigned)ga;                         // global_addr[31:0]
  g0.w = (unsigned)(ga >> 32) | (2u << 30);    // global_addr[56:32] | type=2
  v8i g1;
  g1[0] = (int)(2u << 16);                     // data_size=2 -> 4 bytes
  g1[1] = (int)((tdim0 & 0xffffu) << 16);      // tensor_dim0[15:0]
  g1[2] = (int)((tdim0 >> 16) | ((tdim1 & 0xffffu) << 16));
  g1[3] = (int)((tdim1 >> 16) | (tile0 << 16));// tile_dim0
  g1[4] = (int)tile1;                          // tile_dim1 (tile_dim2=0)
  g1[5] = (int)stride0;                        // tensor_dim0_stride[31:0]
  g1[6] = 0;
  g1[7] = 0;
  v4i z4 = {0, 0, 0, 0};
  v8i z8 = {0, 0, 0, 0, 0, 0, 0, 0};
  __builtin_amdgcn_tensor_load_to_lds(g0, g1, z4, z4, z8, 0);
}
#endif

// ---------------------------------------------------------------------------
// Kernel 1: y = x @ W + bias via V_WMMA_F32_16X16X4_F32 (exact fp32 tensor op)
// Block = 256 threads = 8 waves (wave32). Each wave -> 16(M) x 64(N) strip.
// Double-buffered LDS; tiles staged by the Tensor Data Mover when available.
// ---------------------------------------------------------------------------
__global__ __launch_bounds__(256) void qls_gemm_wmma(
    const float* __restrict__ x, const float* __restrict__ w,
    const float* __restrict__ bias, float* __restrict__ y) {
  __shared__ __attribute__((aligned(16))) float As[2][MT * KC]; // [64][32]
  __shared__ __attribute__((aligned(16))) float Bs[2][KC * NT]; // [32][128]

  const int tid   = threadIdx.x;
  const int lane  = tid & 31;
  const int wid   = tid >> 5;
  const int waveM = wid & 3;   // 0..3  -> 16-row strip
  const int waveN = wid >> 2;  // 0..1  -> 64-col strip
  const int half  = lane >> 4; // 0/1 : K sub-pair select (ISA A/B layout)
  const int l16   = lane & 15;

  const int m_blk = blockIdx.y * MT;
  const int n_blk = blockIdx.x * NT;

  v8f acc[4] = {};   // 4 n-subtiles of 16x16 f32 C/D fragments

#ifdef USE_TDM
  unsigned long long gaA = (unsigned long long)(const void*)(x + (size_t)m_blk * D_MODEL);
  unsigned long long gaB = (unsigned long long)(const void*)(w + n_blk);
  const unsigned ldsA0 = (unsigned)(size_t)(&As[0][0]);
  const unsigned ldsA1 = (unsigned)(size_t)(&As[1][0]);
  const unsigned ldsB0 = (unsigned)(size_t)(&Bs[0][0]);
  const unsigned ldsB1 = (unsigned)(size_t)(&Bs[1][0]);

  // prefetch chunk 0 into buffer 0 (one wave issues; TDM ignores EXEC)
  if (tid < 32) {
    tdm_load_tile_f32(ldsA0, gaA, D_MODEL, NTOK, KC, MT, D_MODEL);
    tdm_load_tile_f32(ldsB0, gaB, NOUT, D_MODEL, NT, KC, NOUT);
  }

  int buf = 0;
  for (int k0 = 0; k0 < D_MODEL; k0 += KC) {
    if (tid < 32) __builtin_amdgcn_s_wait_tensorcnt(0); // current buffer ready
    __syncthreads();                                    // release implies data visible
    if ((k0 + KC < D_MODEL) && (tid < 32)) {            // overlap next DMA w/ compute
      const unsigned long long offA = (unsigned long long)(k0 + KC) * 4ull;
      const unsigned long long offB = (unsigned long long)(k0 + KC) * (unsigned long long)NOUT * 4ull;
      tdm_load_tile_f32(buf ? ldsA0 : ldsA1, gaA + offA, D_MODEL, NTOK, KC, MT, D_MODEL);
      tdm_load_tile_f32(buf ? ldsB0 : ldsB1, gaB + offB, NOUT, D_MODEL, NT, KC, NOUT);
    }

    const float* Abase = &As[buf][(waveM * 16 + l16) * KC];
    const float* Bbase = &Bs[buf][0];
#pragma unroll
    for (int kk = 0; kk < KC; kk += 4) {
      v2f a;                           // A 16x4 frag: VGPR0=K(half?2:0), VGPR1=K+1
      a.x = Abase[kk + half * 2 + 0];
      a.y = Abase[kk + half * 2 + 1];
#pragma unroll
      for (int s = 0; s < 4; ++s) {
        const int n = waveN * 64 + s * 16 + l16;
        v2f b;                         // B 4x16 frag: rows split by `half`
        b.x = Bbase[(kk + half * 2 + 0) * NT + n];
        b.y = Bbase[(kk + half * 2 + 1) * NT + n];
        acc[s] = __builtin_amdgcn_wmma_f32_16x16x4_f32(
            false, a, false, b, (short)0, acc[s], false, false);
      }
    }
    buf ^= 1;
  }
#else
  // fallback: manual global->VGPR->LDS staging, single buffer
  const int a_row = tid >> 2;         // 0..63
  const int a_col = (tid & 3) * 8;    // 0,8,16,24
  const int b_row = tid >> 5;         // 0..7
  const int b_col = (tid & 31) * 4;   // 0..124

  for (int k0 = 0; k0 < D_MODEL; k0 += KC) {
    v4f av0 = *(const v4f*)(x + (size_t)(m_blk + a_row) * D_MODEL + k0 + a_col);
    v4f av1 = *(const v4f*)(x + (size_t)(m_blk + a_row) * D_MODEL + k0 + a_col + 4);
    *(v4f*)(&As[0][a_row * KC + a_col])     = av0;
    *(v4f*)(&As[0][a_row * KC + a_col + 4]) = av1;
#pragma unroll
    for (int r = 0; r < KC; r += 8) {
      v4f bv = *(const v4f*)(w + (size_t)(k0 + b_row + r) * NOUT + n_blk + b_col);
      *(v4f*)(&Bs[0][(b_row + r) * NT + b_col]) = bv;
    }
    __syncthreads();

    const float* Abase = &As[0][(waveM * 16 + l16) * KC];
#pragma unroll
    for (int kk = 0; kk < KC; kk += 4) {
      v2f a;
      a.x = Abase[kk + half * 2 + 0];
      a.y = Abase[kk + half * 2 + 1];
#pragma unroll
      for (int s = 0; s < 4; ++s) {
        const int n = waveN * 64 + s * 16 + l16;
        v2f b;
        b.x = Bs[0][(kk + half * 2 + 0) * NT + n];
        b.y = Bs[0][(kk + half * 2 + 1) * NT + n];
        acc[s] = __builtin_amdgcn_wmma_f32_16x16x4_f32(
            false, a, false, b, (short)0, acc[s], false, false);
      }
    }
    __syncthreads();
  }
#endif

  // write back: C/D fragment layout -> VGPR v holds rows v (lanes 0-15) and v+8
#pragma unroll
  for (int s = 0; s < 4; ++s) {
    const int n  = n_blk + waveN * 64 + s * 16 + l16;
    const float bb = bias[n];
#pragma unroll
    for (int v = 0; v < 8; ++v) {
      const int m = m_blk + waveM * 16 + v + half * 8;
      y[(size_t)m * NOUT + n] = acc[s][v] + bb;
    }
  }
}

// ---------------------------------------------------------------------------
// Kernel 2: per-column order statistics. One block per output channel.
// Column lives in LDS; bitonic sort; exact jnp.quantile('linear'/'lower')
// positions for N=2048; then min/max of where(mask, median, y).
// params[c*8+{0..4}] = lower, upper, median, offset, range
// ---------------------------------------------------------------------------
__global__ __launch_bounds__(256) void qls_colstats(
    const float* __restrict__ y, float* __restrict__ params) {
  __shared__ float s[NTOK];
  __shared__ float red[512];
  const int c   = blockIdx.x;
  const int tid = threadIdx.x;

  for (int i = tid; i < NTOK; i += 256) s[i] = y[(size_t)i * NOUT + c];
  __syncthreads();

  // bitonic sort (n = 2048, power of two; comparator pairs disjoint per stage)
  for (int k = 2; k <= NTOK; k <<= 1) {
    for (int j = k >> 1; j > 0; j >>= 1) {
      for (int i = tid; i < NTOK; i += 256) {
        const int ixj = i ^ j;
        if (ixj > i) {
          const bool up = ((i & k) == 0);
          const float a = s[i], b = s[ixj];
          if ((a > b) == up) { s[i] = b; s[ixj] = a; }
        }
      }
      __syncthreads();
    }
  }

  // quantile positions (match jnp defaults): t = 1 - (1-0.999)/2
  const double t    = 1.0 - (1.0 - 0.999) / 2.0;        // 0.9995
  const double posU = t * (double)(NTOK - 1);           // 2045.9765
  const double posL = (1.0 - t) * (double)(NTOK - 1);   // 1.0235
  const int    fU   = (int)posU;
  const int    fL   = (int)posL;
  const float  upper = (float)((double)s[fU] + (posU - fU) * ((double)s[fU + 1] - (double)s[fU]));
  const float  lower = (float)((double)s[fL] + (posL - fL) * ((double)s[fL + 1] - (double)s[fL]));
  const float  median = s[(NTOK - 1) >> 1];             // method='lower' -> s[1023]

  // min/max of tmp = where(mask, median, y)
  float mn = INFINITY, mx = -INFINITY;
  for (int i = tid; i < NTOK; i += 256) {
    const float v = s[i];
    const float tv = ((v <= lower) || (v >= upper)) ? median : v;
    mn = fminf(mn, tv);
    mx = fmaxf(mx, tv);
  }
  red[tid] = mn; red[256 + tid] = mx;
  __syncthreads();
  for (int off = 128; off > 0; off >>= 1) {
    if (tid < off) {
      red[tid]       = fminf(red[tid],       red[tid + off]);
      red[256 + tid] = fmaxf(red[256 + tid], red[256 + tid + off]);
    }
    __syncthreads();
  }
  if (tid == 0) {
    const float minv = red[0], maxv = red[256];
    float* p = params + (size_t)c * 8;
    p[0] = lower;
    p[1] = upper;
    p[2] = median;
    p[3] = (maxv + minv) * 0.5f;  // offset
    p[4] = (maxv - minv) * 0.5f;  // range
  }
}

// ---------------------------------------------------------------------------
// Kernel 3: NUQ reconstruct: nearest-pole LUT, outlier passthrough, nan_to_num
// ---------------------------------------------------------------------------
__global__ __launch_bounds__(256) void qls_dequant(
    const float* __restrict__ y, const float* __restrict__ params,
    const float* __restrict__ lut, float* __restrict__ out) {
  const size_t i = (size_t)blockIdx.x * 256 + threadIdx.x;
  const int c = (int)(i & (NOUT - 1));
  const float* p = params + (size_t)c * 8;
  const float lower = p[0], upper = p[1], offs = p[3], range = p[4];

  const float v = y[i];
  const bool mask = (v <= lower) || (v >= upper);
  const float inp = v - offs;
  const float outlier = mask ? inp : 0.f;
  const float scaled  = mask ? 0.f : inp / fmaxf(range, EPSV);

  // round_to_nearest_pole_sim: strict '<' keeps first-min (argmin semantics)
  float best = lut[0];
  float bd   = fabsf(scaled - best);
#pragma unroll
  for (int k = 1; k < NPOLES; ++k) {
    const float pk = lut[k];
    const float d  = fabsf(scaled - pk);
    if (d < bd) { bd = d; best = pk; }
  }

  float q = mask ? 0.f : best * range;
  q = q + outlier + offs;
  if (isnan(q) || isinf(q)) q = 0.f;   // nan_to_num(nan=0, posinf=0, neginf=0)
  out[i] = q;
}

// ---------------------------------------------------------------------------
extern "C" void kernel_launch(void* const* d_in, const int* in_sizes, int n_in,
                              void* d_out, int out_size, void* d_ws, size_t ws_size,
                              hipStream_t stream) {
  const float* x    = (const float*)d_in[0];  // (2,1024,4096)
  const float* w    = (const float*)d_in[1];  // (4096,4096) = W^T storage
  const float* bias = (const float*)d_in[2];  // (4096,)
  const float* lut  = (const float*)d_in[3];  // (16,)
  float* out = (float*)d_out;                 // (2,1024,4096)

  float* y      = (float*)d_ws;                        // 2048*4096 f32 = 32 MB
  float* params = y + (size_t)NTOK * NOUT;             // 4096*8 f32

  dim3 g1(NOUT / NT, NTOK / MT);                       // 32 x 32 macro-tiles
  qls_gemm_wmma<<<g1, 256, 0, stream>>>(x, w, bias, y);

  qls_colstats<<<NOUT, 256, 0, stream>>>(y, params);

  const int total = NTOK * NOUT;
  qls_dequant<<<total / 256, 256, 0, stream>>>(y, params, lut, out);
}